// SparseBackbone3D_45784351375625
// MI455X (gfx1250) — compile-verified
//
#include <hip/hip_runtime.h>
#include <hip/hip_bf16.h>
#include <stdint.h>

typedef float v2f __attribute__((ext_vector_type(2)));
typedef float v8f __attribute__((ext_vector_type(8)));

#define CDIM 128
#define HASH_P (1u << 20)
#define EMPTY_KEY 0xFFFFFFFFFFFFFFFFull

// ---------------------------------------------------------------------------
// Stage 1: Y = relu(X) @ W   via V_WMMA_F32_16X16X4_F32, W staged in LDS.
// Block = 256 threads = 8 waves; each wave computes a 16-row strip of Y.
// ---------------------------------------------------------------------------
__global__ __launch_bounds__(256) void k_gemm_relu_wmma(
    const float* __restrict__ X, const float* __restrict__ W,
    float* __restrict__ Y, int M)
{
  __shared__ float lw[CDIM * CDIM];   // 64 KB of the 320 KB WGP LDS
  const int tid = threadIdx.x;
  {
    const float4* W4 = (const float4*)W;
    float4* L4 = (float4*)lw;
#pragma unroll
    for (int i = 0; i < (CDIM * CDIM / 4) / 256; ++i)
      L4[tid + i * 256] = W4[tid + i * 256];
  }
  __syncthreads();

  const int wave = tid >> 5;
  const int lane = tid & 31;
  const int half = lane >> 4;     // 0: K+0/K+1 (rows M+0..7 of D), 1: K+2/K+3 (rows M+8..15)
  const int r    = lane & 15;
  const long m0  = ((long)blockIdx.x * 8 + wave) * 16;
  if (m0 + 16 > (long)M) return;

  // A fragments for all 32 K-steps (relu folded into the load).
  v2f a[32];
  const float* Ar = X + (m0 + r) * CDIM;
#pragma unroll
  for (int kk = 0; kk < 32; ++kk) {
    const int c = kk * 4 + half * 2;
    a[kk].x = fmaxf(Ar[c],     0.0f);
    a[kk].y = fmaxf(Ar[c + 1], 0.0f);
  }

  for (int n = 0; n < 8; ++n) {
    v8f acc = {};
    const float* Bc = lw + n * 16 + r;
#pragma unroll
    for (int kk = 0; kk < 32; ++kk) {
      const int k0 = kk * 4 + half * 2;
      v2f b;
      b.x = Bc[(k0 + 0) * CDIM];
      b.y = Bc[(k0 + 1) * CDIM];
      acc = __builtin_amdgcn_wmma_f32_16x16x4_f32(
          /*neg_a=*/false, a[kk], /*neg_b=*/false, b,
          /*c_mod=*/(short)0, acc, /*reuse_a=*/false, /*reuse_b=*/false);
    }
    // D layout: VGPR v -> row m0 + half*8 + v, col n*16 + r
    float* Yp = Y + (m0 + half * 8) * CDIM + n * 16 + r;
#pragma unroll
    for (int v = 0; v < 8; ++v)
      Yp[v * CDIM] = acc[v];
  }
}

// ---------------------------------------------------------------------------
// Utility kernels
// ---------------------------------------------------------------------------
__global__ void k_zero4(float4* __restrict__ p, long n4) {
  long i = (long)blockIdx.x * blockDim.x + threadIdx.x;
  if (i < n4) p[i] = make_float4(0.f, 0.f, 0.f, 0.f);
}

__global__ void k_relu4(float4* __restrict__ p, long n4) {
  long i = (long)blockIdx.x * blockDim.x + threadIdx.x;
  if (i < n4) {
    float4 v = p[i];
    v.x = fmaxf(v.x, 0.f); v.y = fmaxf(v.y, 0.f);
    v.z = fmaxf(v.z, 0.f); v.w = fmaxf(v.w, 0.f);
    p[i] = v;
  }
}

__global__ void k_init_table(unsigned long long* __restrict__ keys,
                             int* __restrict__ uids,
                             int* __restrict__ counter, int P) {
  int i = blockIdx.x * blockDim.x + threadIdx.x;
  if (i < P) { keys[i] = EMPTY_KEY; uids[i] = -1; }
  if (i == 0) *counter = 0;
}

__device__ __forceinline__ uint64_t mix64(uint64_t k) {
  k *= 0x9E3779B97F4A7C15ull;
  k ^= k >> 29; k *= 0xBF58476D1CE4E5B9ull;
  k ^= k >> 32;
  return k;
}

// ---------------------------------------------------------------------------
// Stage 2a: quantize coords (>>1), dedup via hash table, emit inverse indices
// and the unique (quantized) coords for the next layer.
// ---------------------------------------------------------------------------
__global__ void k_dedup(const int* __restrict__ coords_in, int n_fixed,
                        const int* __restrict__ n_ptr,
                        unsigned long long* __restrict__ tkeys,
                        int* __restrict__ tuids,
                        int* __restrict__ counter,
                        int* __restrict__ inv,
                        int* __restrict__ uniq_coords)
{
  const int n = n_ptr ? *n_ptr : n_fixed;
  const int i = blockIdx.x * blockDim.x + threadIdx.x;
  if (i >= n) return;

  const int b = coords_in[i * 4 + 0] >> 1;
  const int z = coords_in[i * 4 + 1] >> 1;
  const int y = coords_in[i * 4 + 2] >> 1;
  const int x = coords_in[i * 4 + 3] >> 1;
  const uint64_t key = ((uint64_t)(uint32_t)b << 45) |
                       ((uint64_t)(uint32_t)z << 30) |
                       ((uint64_t)(uint32_t)y << 15) |
                        (uint64_t)(uint32_t)x;

  uint32_t h = (uint32_t)mix64(key) & (HASH_P - 1);
  int uid = -1;
  for (;;) {
    unsigned long long prev =
        atomicCAS(&tkeys[h], (unsigned long long)EMPTY_KEY, (unsigned long long)key);
    if (prev == EMPTY_KEY) {               // this lane claims the slot
      uid = atomicAdd(counter, 1);
      uniq_coords[uid * 4 + 0] = b;
      uniq_coords[uid * 4 + 1] = z;
      uniq_coords[uid * 4 + 2] = y;
      uniq_coords[uid * 4 + 3] = x;
      __hip_atomic_store(&tuids[h], uid, __ATOMIC_RELEASE, __HIP_MEMORY_SCOPE_AGENT);
      break;
    } else if (prev == (unsigned long long)key) {   // slot owned by same key
      while ((uid = __hip_atomic_load(&tuids[h], __ATOMIC_ACQUIRE,
                                      __HIP_MEMORY_SCOPE_AGENT)) < 0) { }
      break;
    }
    h = (h + 1) & (HASH_P - 1);            // linear probe
  }
  inv[i] = uid;
}

// ---------------------------------------------------------------------------
// Stage 2b: segment-sum scatter-add (float atomics), optional relu on input.
// One thread per (row, 4-float chunk): 32 chunks per 128-wide row.
// ---------------------------------------------------------------------------
__global__ void k_scatter(const float* __restrict__ fin,
                          const int* __restrict__ inv,
                          float* __restrict__ fout,
                          int n_fixed, const int* __restrict__ n_ptr,
                          int relu_in)
{
  const int n = n_ptr ? *n_ptr : n_fixed;
  long t = (long)blockIdx.x * blockDim.x + threadIdx.x;
  long i = t >> 5;
  int  c = (int)(t & 31) * 4;
  if (i >= (long)n) return;

  float4 v = *(const float4*)(fin + i * CDIM + c);
  if (relu_in) {
    v.x = fmaxf(v.x, 0.f); v.y = fmaxf(v.y, 0.f);
    v.z = fmaxf(v.z, 0.f); v.w = fmaxf(v.w, 0.f);
  }
  float* dst = fout + (long)inv[i] * CDIM + c;
  atomicAdd(dst + 0, v.x);
  atomicAdd(dst + 1, v.y);
  atomicAdd(dst + 2, v.z);
  atomicAdd(dst + 3, v.w);
}

// ---------------------------------------------------------------------------
// Host-side orchestration
// ---------------------------------------------------------------------------
extern "C" void kernel_launch(void* const* d_in, const int* in_sizes, int n_in,
                              void* d_out, int out_size, void* d_ws, size_t ws_size,
                              hipStream_t stream) {
  (void)n_in; (void)out_size; (void)ws_size;
  const float* X  = (const float*)d_in[0];   // (M,128) voxel_features
  const int*   CO = (const int*)d_in[1];     // (M,4)   voxel_coords
  const float* W0 = (const float*)d_in[2];   // (128,128)
  const int M = in_sizes[0] / CDIM;

  // Workspace carve-up
  char* p = (char*)d_ws;
  const size_t featsBytes = (size_t)M * CDIM * sizeof(float);
  float* featsA = (float*)p;                 p += featsBytes;            // GEMM out
  float* featsB = (float*)p;                 p += featsBytes;            // layer-2 out
  unsigned long long* tkeys = (unsigned long long*)p; p += (size_t)HASH_P * 8;
  int* tuids  = (int*)p;                     p += (size_t)HASH_P * 4;
  int* inv    = (int*)p;                     p += (size_t)M * 4;
  int* uniqcA = (int*)p;                     p += (size_t)M * 16;
  int* uniqcB = (int*)p;                     p += (size_t)M * 16;
  int* counters = (int*)p;                   p += 256;                   // [0]=cnt2 [1]=cnt3
  float* OUT = (float*)d_out;

  const long n4 = (long)M * CDIM / 4;
  const int  ZB = (int)((n4 + 255) / 256);
  const int  TB = (HASH_P + 255) / 256;
  const int  MB = (M + 255) / 256;
  const long scatterT = (long)M * 32;
  const int  SB = (int)((scatterT + 255) / 256);

  // Zero accumulation targets (harness poisons d_out/d_ws).
  k_zero4<<<ZB, 256, 0, stream>>>((float4*)featsB, n4);
  k_zero4<<<ZB, 256, 0, stream>>>((float4*)OUT,    n4);

  // Layer 1: feats = relu(X) @ W0   (M/128 blocks of 8 waves x 16 rows)
  k_gemm_relu_wmma<<<(M + 127) / 128, 256, 0, stream>>>(X, W0, featsA, M);

  // Layer 2: quantize -> unique -> segment_sum
  k_init_table<<<TB, 256, 0, stream>>>(tkeys, tuids, &counters[0], (int)HASH_P);
  k_dedup<<<MB, 256, 0, stream>>>(CO, M, nullptr, tkeys, tuids, &counters[0],
                                  inv, uniqcA);
  k_scatter<<<SB, 256, 0, stream>>>(featsA, inv, featsB, M, nullptr, /*relu_in=*/0);

  // Layer 3: quantize uniq coords -> unique -> segment_sum(relu(featsB))
  k_init_table<<<TB, 256, 0, stream>>>(tkeys, tuids, &counters[1], (int)HASH_P);
  k_dedup<<<MB, 256, 0, stream>>>(uniqcA, M, &counters[0], tkeys, tuids,
                                  &counters[1], inv, uniqcB);
  k_scatter<<<SB, 256, 0, stream>>>(featsB, inv, OUT, M, &counters[0], /*relu_in=*/1);

  // Final relu on the output.
  k_relu4<<<ZB, 256, 0, stream>>>((float4*)OUT, n4);
}